// GraphConvLayer_14147622273528
// MI455X (gfx1250) — compile-verified
//
#include <hip/hip_runtime.h>

typedef float v2f __attribute__((ext_vector_type(2)));
typedef float v8f __attribute__((ext_vector_type(8)));

#define N_NODES 100000
#define N_EDGES 1600000
#define DIM     128

// ---------------------------------------------------------------------------
// Kernel 0: zero workspace (agg [N*D] + deg [N]) with float4 stores.
// ---------------------------------------------------------------------------
__global__ void __launch_bounds__(256) gc_zero_kernel(float4* __restrict__ p, int n4) {
    int i = blockIdx.x * blockDim.x + threadIdx.x;
    if (i < n4) {
        float4 z; z.x = 0.f; z.y = 0.f; z.z = 0.f; z.w = 0.f;
        p[i] = z;
    }
}

// ---------------------------------------------------------------------------
// Kernel 1: edge scatter-add. One wave (32 lanes) per edge.
// Each lane loads float4 of x[src] (coalesced 512B per edge) and does
// 4 global_atomic_add_f32 into agg[dst]. Lane 0 bumps deg[dst].
// ---------------------------------------------------------------------------
__global__ void __launch_bounds__(256) gc_agg_kernel(const float* __restrict__ x,
                                                     const long long* __restrict__ ei,
                                                     float* __restrict__ agg,
                                                     float* __restrict__ deg,
                                                     int nEdges) {
    int gid  = blockIdx.x * blockDim.x + threadIdx.x;
    int edge = gid >> 5;
    int lane = gid & 31;
    if (edge >= nEdges) return;

    long long src = ei[edge];              // row 0 of edge_index
    long long dst = ei[(long long)nEdges + edge];  // row 1 of edge_index

    const float4* xs = (const float4*)(x + src * DIM);
    float4 v = xs[lane];                   // 32 lanes * 4 floats = 128 features

    float* ad = agg + dst * DIM + lane * 4;
    atomicAdd(ad + 0, v.x);
    atomicAdd(ad + 1, v.y);
    atomicAdd(ad + 2, v.z);
    atomicAdd(ad + 3, v.w);

    if (lane == 0) atomicAdd(deg + dst, 1.0f);
}

// ---------------------------------------------------------------------------
// Kernel 2: fused dual GEMM + bias + ReLU using V_WMMA_F32_16X16X4_F32.
// Block = 256 threads = 8 waves. Block b handles node rows [16b, 16b+16);
// wave w handles output cols [16w, 16w+16). Full f32 precision.
//
// A fragment (16x4 f32, 2 VGPRs): lane L holds A[M=L%16][K=2*(L/16)+{0,1}].
// B fragment (4x16 f32, 2 VGPRs): lane L holds B[K=2*(L/16)+{0,1}][N=L%16];
//   B = W^T so B[k][n] = W[col0+n][k] -> contiguous float2 along W's k axis.
// C/D (16x16 f32, 8 VGPRs): VGPR v -> row v (lanes 0-15), row v+8 (lanes 16-31).
// ---------------------------------------------------------------------------
__global__ void __launch_bounds__(256) gc_gemm_kernel(const float* __restrict__ x,
                                                      const float* __restrict__ agg,
                                                      const float* __restrict__ deg,
                                                      const float* __restrict__ Wself,
                                                      const float* __restrict__ bself,
                                                      const float* __restrict__ Wneigh,
                                                      const float* __restrict__ bneigh,
                                                      float* __restrict__ out) {
    const int wave  = threadIdx.x >> 5;     // 0..7 -> col tile
    const int lane  = threadIdx.x & 31;
    const int node0 = blockIdx.x * 16;
    const int col0  = wave * 16;

    const int mrow  = lane & 15;            // M (A rows) / N (B cols) within tile
    const int khalf = (lane >> 4) * 2;      // 0 or 2

    const int arow = node0 + mrow;          // node row this lane loads for A
    const int wrow = col0 + mrow;           // weight row this lane loads for B

    v8f c = {};                             // f32 accumulator, starts at 0

    // ---- x @ W_self^T ----
    {
        const float* ap = x     + (long)arow * DIM + khalf;
        const float* bp = Wself + (long)wrow * DIM + khalf;
        #pragma unroll
        for (int k = 0; k < DIM; k += 4) {
            v2f a = *(const v2f*)(ap + k);
            v2f b = *(const v2f*)(bp + k);
            c = __builtin_amdgcn_wmma_f32_16x16x4_f32(
                    false, a, false, b, (short)0, c, false, false);
        }
    }

    // ---- (agg/deg) @ W_neigh^T, normalization fused into the A load ----
    {
        float d  = deg[arow];
        float rd = 1.0f / fmaxf(d, 1.0f);
        const float* ap = agg    + (long)arow * DIM + khalf;
        const float* bp = Wneigh + (long)wrow * DIM + khalf;
        #pragma unroll
        for (int k = 0; k < DIM; k += 4) {
            v2f a = *(const v2f*)(ap + k);
            a.x *= rd; a.y *= rd;
            v2f b = *(const v2f*)(bp + k);
            c = __builtin_amdgcn_wmma_f32_16x16x4_f32(
                    false, a, false, b, (short)0, c, false, false);
        }
    }

    // ---- bias + ReLU + store ----
    const int   ocol = col0 + (lane & 15);
    const float bias = bself[ocol] + bneigh[ocol];
    const int   rbase = node0 + ((lane >> 4) << 3);   // +8 for upper half-wave
    #pragma unroll
    for (int v = 0; v < 8; ++v) {
        float val = c[v] + bias;
        val = val > 0.0f ? val : 0.0f;
        out[(long)(rbase + v) * DIM + ocol] = val;
    }
}

// ---------------------------------------------------------------------------
// Launch
// ---------------------------------------------------------------------------
extern "C" void kernel_launch(void* const* d_in, const int* in_sizes, int n_in,
                              void* d_out, int out_size, void* d_ws, size_t ws_size,
                              hipStream_t stream) {
    const float*     x   = (const float*)d_in[0];
    const long long* ei  = (const long long*)d_in[1];   // int64 edge_index [2, E]
    const float*     Ws  = (const float*)d_in[2];
    const float*     bs  = (const float*)d_in[3];
    const float*     Wn  = (const float*)d_in[4];
    const float*     bn  = (const float*)d_in[5];
    float*           out = (float*)d_out;

    // workspace: agg [N*D] floats, then deg [N] floats  (~51.6 MB)
    float* agg = (float*)d_ws;
    float* deg = agg + (size_t)N_NODES * DIM;

    // 0) zero agg + deg  (N*D + N = 12,900,000 floats, divisible by 4)
    {
        int n4 = (N_NODES * DIM + N_NODES) / 4;
        int blocks = (n4 + 255) / 256;
        gc_zero_kernel<<<blocks, 256, 0, stream>>>((float4*)d_ws, n4);
    }

    // 1) edge scatter-add: 8 edges per 256-thread block
    {
        int blocks = N_EDGES / 8;   // 1.6M divisible by 8
        gc_agg_kernel<<<blocks, 256, 0, stream>>>(x, ei, agg, deg, N_EDGES);
    }

    // 2) fused dual WMMA GEMM + bias + ReLU: 16 node rows per block
    {
        int blocks = N_NODES / 16;  // 100000 / 16 = 6250 exactly
        gc_gemm_kernel<<<blocks, 256, 0, stream>>>(x, agg, deg, Ws, bs, Wn, bn, out);
    }
}